// ScaledDotProductAttention_Operator_19181323944265
// MI455X (gfx1250) — compile-verified
//
#include <hip/hip_runtime.h>

// CDNA5 / gfx1250 fused attention (fp32, WMMA 16x16x4 f32).
//
// Roofline: ~268 MB HBM traffic, 4.3 GFLOP -> memory-bound (~11.5us @ 23.3TB/s).
// fp32 WMMA keeps full precision and removes VALU from the critical path.

typedef __attribute__((ext_vector_type(2))) float v2f;
typedef __attribute__((ext_vector_type(8))) float v8f;

#define F_DIM      16384            // x*y*d = 32*32*16
#define P_DIM      64
#define BN_TOT     16               // b*n = 2*8
#define KSPLIT     8
#define KCHUNK     (F_DIM / KSPLIT) // 2048
#define OUT_ELEMS  (2 * 64 * 32 * 32 * 16 * 8) // 16777216
#define INV_SCALE  (1.0f / 1024.0f)

// ---------------------------------------------------------------------------
// K1: partial scores.  grid = 512 blocks = (bn=16) x (ptile=4) x (ksplit=8),
// 128 threads = 4 waves, one 16x16 tile per wave (qtile = wave id).
// WMMA f32 16x16x4:  A[m][k] lane=m(0..15), k = 2*(lane/16)+j  (float2 load)
//                    B[k][n] lane=n(0..15), k = 2*(lane/16)+j  (float2 load)
// ---------------------------------------------------------------------------
__global__ void attn_scores_partial(const float* __restrict__ Q,
                                    const float* __restrict__ K,
                                    float* __restrict__ part) {
    const int bx   = blockIdx.x;
    const int ks   = bx & 7;
    const int pt   = (bx >> 3) & 3;
    const int bn   = bx >> 5;
    const int tid  = threadIdx.x;
    const int qt   = tid >> 5;   // wave id 0..3 -> q tile
    const int lane = tid & 31;
    const int m    = lane & 15;
    const int half = lane >> 4;

    const float* qptr = Q + (size_t)(bn * 64 + pt * 16 + m) * F_DIM
                          + (size_t)(ks * KCHUNK) + 2 * half;
    const float* kptr = K + (size_t)(bn * 64 + qt * 16 + m) * F_DIM
                          + (size_t)(ks * KCHUNK) + 2 * half;

    v8f c = {};
#pragma unroll 4
    for (int t = 0; t < KCHUNK; t += 4) {
        v2f a = *(const v2f*)(qptr + t);
        v2f b = *(const v2f*)(kptr + t);
        // 8 args: (neg_a, A, neg_b, B, c_mod, C, reuse_a, reuse_b)
        c = __builtin_amdgcn_wmma_f32_16x16x4_f32(false, a, false, b,
                                                  (short)0, c, false, false);
    }

    // D layout: VGPR r -> M = r + 8*half, N = lane%16
    const int qcol = qt * 16 + m;
#pragma unroll
    for (int r = 0; r < 8; ++r) {
        const int p = pt * 16 + r + 8 * half;
        part[(size_t)(((bn * 8 + ks) * 64) + p) * 64 + qcol] = c[r];
    }
}

// ---------------------------------------------------------------------------
// K2: reduce K-split partials + scale + row softmax.  grid = 16, 64 threads;
// thread p owns score row p of its (b,n).
// ---------------------------------------------------------------------------
__global__ void attn_softmax(const float* __restrict__ part,
                             float* __restrict__ attn) {
    const int bn = blockIdx.x;
    const int p  = threadIdx.x;          // 0..63
    __shared__ float sh[64][65];         // stride 65 -> conflict-free rows

    float mx = -3.402823466e38f;
    for (int q = 0; q < 64; ++q) {
        float s = 0.0f;
#pragma unroll
        for (int ks = 0; ks < 8; ++ks)
            s += part[(size_t)(((bn * 8 + ks) * 64) + p) * 64 + q];
        s *= INV_SCALE;
        sh[p][q] = s;
        mx = fmaxf(mx, s);
    }
    float sum = 0.0f;
    for (int q = 0; q < 64; ++q) {
        const float e = __expf(sh[p][q] - mx);
        sh[p][q] = e;
        sum += e;
    }
    const float inv = 1.0f / sum;
    for (int q = 0; q < 64; ++q)
        attn[(size_t)(bn * 64 + p) * 64 + q] = sh[p][q] * inv;
}

// ---------------------------------------------------------------------------
// K3: out = attn x V via WMMA, writing the [b,p,x,y,d,n] transpose.
// grid = 4096 blocks = (bn=16) x (fchunk=256 of 64 features), 512 threads
// = 16 waves = 4 ptiles x 4 ftiles; each wave: 16 wmma iters over q (K=64).
// ---------------------------------------------------------------------------
__global__ void attn_output(const float* __restrict__ ATT,
                            const float* __restrict__ V,
                            float* __restrict__ out) {
    const int bx   = blockIdx.x;
    const int bn   = bx >> 8;    // 0..15
    const int fc   = bx & 255;   // 0..255
    const int tid  = threadIdx.x;
    const int w    = tid >> 5;   // 0..15
    const int lane = tid & 31;
    const int pt   = w >> 2;     // 0..3
    const int ft   = w & 3;      // 0..3
    const int m    = lane & 15;
    const int half = lane >> 4;
    const int f    = fc * 64 + ft * 16 + m;

    const float* aptr  = ATT + (size_t)(bn * 64 + pt * 16 + m) * 64 + 2 * half;
    const float* vbase = V   + (size_t)(bn * 64) * F_DIM + f;

    v8f c = {};
#pragma unroll
    for (int t = 0; t < 16; ++t) {
        v2f a = *(const v2f*)(aptr + 4 * t);          // attn row, contiguous q
        const int q = 4 * t + 2 * half;
        v2f b;
        b.x = vbase[(size_t)q * F_DIM];               // v[q  ][f], lanes coalesced in f
        b.y = vbase[(size_t)(q + 1) * F_DIM];         // v[q+1][f]
        c = __builtin_amdgcn_wmma_f32_16x16x4_f32(false, a, false, b,
                                                  (short)0, c, false, false);
    }

    const int b_idx = bn >> 3;
    const int n     = bn & 7;
#pragma unroll
    for (int r = 0; r < 8; ++r) {
        const int p = pt * 16 + r + 8 * half;
        // out[b][p][f][n], n innermost (stride 8)
        out[((size_t)(b_idx * 64 + p) * F_DIM + f) * 8 + n] = c[r];
    }
}

extern "C" void kernel_launch(void* const* d_in, const int* in_sizes, int n_in,
                              void* d_out, int out_size, void* d_ws, size_t ws_size,
                              hipStream_t stream) {
    (void)in_sizes; (void)n_in; (void)out_size; (void)d_ws; (void)ws_size;
    const float* Q  = (const float*)d_in[0];
    const float* Kk = (const float*)d_in[1];
    const float* V  = (const float*)d_in[2];
    float* out  = (float*)d_out;
    float* part = out;              // scratch: out region is overwritten by K3
    float* attn = out + OUT_ELEMS;  // final attention_weights region

    attn_scores_partial<<<dim3(512),  dim3(128), 0, stream>>>(Q, Kk, part);
    attn_softmax       <<<dim3(16),   dim3(64),  0, stream>>>(part, attn);
    attn_output        <<<dim3(4096), dim3(512), 0, stream>>>(attn, V, out);
}